// GMMWeightedCond_63745904607832
// MI455X (gfx1250) — compile-verified
//
#include <hip/hip_runtime.h>

#define NOUT 11
#define INV_PI     0.31830988618f
#define PI_OVER_2  1.57079632679f
#define PI_OVER_4  0.78539816339f
#define TWO_PI     6.28318530718f
#define NEG_LOG_2PI -1.8378770664093453f   // -0.5*DIM*log(2*pi), DIM=2

typedef __attribute__((ext_vector_type(2))) float v2f;
typedef __attribute__((ext_vector_type(8))) float v8f;

__device__ __forceinline__ v8f wmma_f32_k4(v2f a, v2f b, v8f c) {
  // D = A(16x4,f32) * B(4x16,f32) + C(16x16,f32)
  return __builtin_amdgcn_wmma_f32_16x16x4_f32(false, a, false, b, (short)0, c,
                                               false, false);
}

__global__ __launch_bounds__(256) void gmm_prior_sample_kernel(
    const float* __restrict__ X,    // N x 6
    const float* __restrict__ RS,   // N x 2
    const float* __restrict__ W1,   // 6 x 32
    const float* __restrict__ B1,   // 32
    const float* __restrict__ W2,   // 32 x 11
    const float* __restrict__ B2,   // 11
    float* __restrict__ outZ,       // N x 2
    float* __restrict__ outLP,      // N
    int N)
{
  // Per-wave private LDS regions (same-wave LDS ops are in-order; no barriers).
  __shared__ float hbuf[8][2][16 * 32];   // H tiles, row-major (32 KB)
  __shared__ float dbuf[8][32 * 12];      // per-sample MLP outputs (12 KB)

  const int lane = threadIdx.x & 31;
  const int wave = threadIdx.x >> 5;
  const int half = lane >> 4;   // lane-half selects K pair (A/B) or M+8 (C/D)
  const int lo   = lane & 15;
  const long base = ((long)blockIdx.x * 8 + wave) * 32;  // first sample of chunk
  const long Nl = (long)N;

  // ---------- uniform fragments: W1 / W2 / biases ----------
  // B-layout 4x16: VGPR v, lane L -> row k = 4j + v + 2*(L>>4), col n = L&15
  v2f b1f[2][2];
#pragma unroll
  for (int j = 0; j < 2; ++j)
#pragma unroll
    for (int nt = 0; nt < 2; ++nt) {
      int n  = lo + 16 * nt;
      int k0 = 4 * j + 2 * half;
      b1f[j][nt].x = (k0     < 6) ? W1[k0 * 32 + n]       : 0.0f;
      b1f[j][nt].y = (k0 + 1 < 6) ? W1[(k0 + 1) * 32 + n] : 0.0f;
    }
  float bias1[2] = { B1[lo], B1[lo + 16] };

  v2f b2f[8];
#pragma unroll
  for (int j = 0; j < 8; ++j) {
    int k0 = 4 * j + 2 * half;
    b2f[j].x = (lo < NOUT) ? W2[k0 * NOUT + lo]       : 0.0f;
    b2f[j].y = (lo < NOUT) ? W2[(k0 + 1) * NOUT + lo] : 0.0f;
  }
  float bias2 = (lo < NOUT) ? B2[lo] : 0.0f;

  // ---------- load X tiles in A-layout (16x4 f32: m=L&15, k=v+2*(L>>4)) ----------
  v2f a[2][2];
#pragma unroll
  for (int t = 0; t < 2; ++t) {
    long row = base + t * 16 + lo;
    if (row >= Nl) row = Nl - 1;               // clamp (tail-safe)
    const float* xr = X + row * 6;
#pragma unroll
    for (int j = 0; j < 2; ++j) {
      int f0 = 4 * j + 2 * half;
      a[t][j].x = (f0     < 6) ? xr[f0]     : 0.0f;
      a[t][j].y = (f0 + 1 < 6) ? xr[f0 + 1] : 0.0f;
    }
  }

  // ---------- GEMM1: H = relu(X@W1 + b1), write H row-major to LDS ----------
#pragma unroll
  for (int t = 0; t < 2; ++t) {
#pragma unroll
    for (int nt = 0; nt < 2; ++nt) {
      v8f c = {};
      c = wmma_f32_k4(a[t][0], b1f[0][nt], c);
      c = wmma_f32_k4(a[t][1], b1f[1][nt], c);
#pragma unroll
      for (int r = 0; r < 8; ++r) {
        int m = r + 8 * half;                  // row (sample in tile)
        int n = lo + 16 * nt;                  // col (hidden unit)
        hbuf[wave][t][m * 32 + n] = fmaxf(c[r] + bias1[nt], 0.0f);
      }
    }
  }

  // ---------- GEMM2: D = H@W2 (K=32 -> 8 k-steps), A-frags gathered from LDS ----------
#pragma unroll
  for (int t = 0; t < 2; ++t) {
    v8f d = {};
#pragma unroll
    for (int j = 0; j < 8; ++j) {
      int k0 = 4 * j + 2 * half;
      v2f a2;
      a2.x = hbuf[wave][t][lo * 32 + k0];
      a2.y = hbuf[wave][t][lo * 32 + k0 + 1];
      d = wmma_f32_k4(a2, b2f[j], d);
    }
    if (lo < NOUT) {                            // EXEC restored before next WMMA
#pragma unroll
      for (int r = 0; r < 8; ++r) {
        int m = r + 8 * half;                   // sample within tile
        dbuf[wave][(t * 16 + m) * 12 + lo] = d[r] + bias2;
      }
    }
  }

  // ---------- elementwise: lane L owns sample base+L ----------
  float v[NOUT];
#pragma unroll
  for (int o = 0; o < NOUT; ++o) v[o] = dbuf[wave][lane * 12 + o];

  long i = base + lane;
  long iclamp = (i < Nl) ? i : (Nl - 1);
  float2 rs = *(const float2*)(RS + iclamp * 2);
  float rdn = rs.x, u1 = rs.y;

  // mixture weights
  float w0 = fabsf(v[8]), w1 = fabsf(v[9]), w2 = fabsf(v[10]);
  float inv = 1.0f / (w0 + w1 + w2);
  w0 *= inv; w1 *= inv; w2 *= inv;
  float wc0 = w0, wc1 = w0 + w1;

  bool g1 = rdn < wc0;
  bool g2 = (!g1) && (rdn < wc1);
  bool gauss = g1 || g2;

  float u0 = g1 ? (rdn / wc0) : (g2 ? (rdn - wc0) / w1 : (rdn - wc1) / w2);
  float U1 = gauss ? u0 : 0.5f;
  float R = sqrtf(-2.0f * __logf(U1));
  float st, ct;
  __sincosf(TWO_PI * u1, &st, &ct);
  float eg0 = R * ct, eg1 = R * st;

  float loc0 = g2 ? v[2] : v[0];
  float loc1 = g2 ? v[3] : v[1];
  float ls0  = g2 ? v[6] : v[4];
  float ls1  = g2 ? v[7] : v[5];
  float zg0 = eg0 * __expf(ls0) + loc0;
  float zg1 = eg1 * __expf(ls1) + loc1;

  // lambert branch
  float wo0 = u0 * 2.0f - 1.0f, wo1 = u1 * 2.0f - 1.0f;
  bool nonzero = !((wo0 == 0.0f) && (wo1 == 0.0f));
  bool c1 = (fabsf(wo0) > fabsf(wo1)) && nonzero;
  bool c2 = (!c1) && nonzero;
  float s0 = (wo0 == 0.0f) ? 1.0f : wo0;
  float s1 = (wo1 == 0.0f) ? 1.0f : wo1;
  float sa1, ca1, sa2, ca2;
  __sincosf(PI_OVER_4 * wo1 / s0, &sa1, &ca1);
  __sincosf(PI_OVER_2 - PI_OVER_4 * wo0 / s1, &sa2, &ca2);
  float zl0 = c1 ? wo0 * ca1 : (c2 ? wo1 * ca2 : 0.0f);
  float zl1 = c1 ? wo0 * sa1 : (c2 ? wo1 * sa2 : 0.0f);

  float z0 = gauss ? zg0 : zl0;
  float z1 = gauss ? zg1 : zl1;

  // log-prob (logsumexp over 2 gaussians + lambert)
  float lg[2];
#pragma unroll
  for (int m = 0; m < 2; ++m) {
    float lsa = v[4 + m * 2], lsb = v[5 + m * 2];
    float e0 = (z0 - v[m * 2])     * __expf(-lsa);
    float e1 = (z1 - v[m * 2 + 1]) * __expf(-lsb);
    float wm = m ? w1 : w0;
    lg[m] = NEG_LOG_2PI + __logf(wm + 1e-5f)
          - 0.5f * (e0 * e0 + e1 * e1) - (lsa + lsb);
  }
  bool invalid = (z0 * z0 + z1 * z1) > 1.0f;
  float pl = invalid ? 0.0f : INV_PI;
  float llp = __logf(pl + 1e-5f) + __logf(w2);

  float mx = fmaxf(fmaxf(lg[0], lg[1]), llp);
  float lp = mx + __logf(__expf(lg[0] - mx) + __expf(lg[1] - mx) + __expf(llp - mx));

  if (i < Nl) {
    outZ[i * 2 + 0] = z0;
    outZ[i * 2 + 1] = z1;
    outLP[i] = lp;
  }
}

extern "C" void kernel_launch(void* const* d_in, const int* in_sizes, int n_in,
                              void* d_out, int out_size, void* d_ws, size_t ws_size,
                              hipStream_t stream) {
  const float* X  = (const float*)d_in[0];
  const float* RS = (const float*)d_in[1];
  const float* W1 = (const float*)d_in[2];
  const float* B1 = (const float*)d_in[3];
  const float* W2 = (const float*)d_in[4];
  const float* B2 = (const float*)d_in[5];
  int N = in_sizes[1] / 2;          // randseed is N x 2
  float* outZ  = (float*)d_out;
  float* outLP = outZ + (long)N * 2;
  int grid = (N + 255) / 256;       // 256 threads = 8 waves, 32 samples/wave
  gmm_prior_sample_kernel<<<grid, 256, 0, stream>>>(X, RS, W1, B1, W2, B2,
                                                    outZ, outLP, N);
}